// SubsampleGroup_1468878815318
// MI455X (gfx1250) — compile-verified
//
#include <hip/hip_runtime.h>
#include <hip/hip_bf16.h>

// Problem constants (from reference)
#define BATCH 8
#define NPTS 16384
#define NCH 64
#define MGRP 1024      // NUM_GROUPS
#define KGRP 32        // GROUP_SIZE
#define RAD2 0.01f     // RADIUS^2
#define NCHUNK (NPTS / 16)   // 1024 chunks of 16 points

typedef __attribute__((ext_vector_type(2))) float v2f;
typedef __attribute__((ext_vector_type(8))) float v8f;

// ---------------------------------------------------------------------------
// Kernel 1: furthest point sampling. One block per batch, dist[] in registers.
// 1024 threads -> 16 points/thread -> short per-iteration critical path.
// ---------------------------------------------------------------------------
#define FPS_T 1024
#define FPS_PPT (NPTS / FPS_T)   // 16 points per thread

__global__ void __launch_bounds__(FPS_T) fps_kernel(const float* __restrict__ p,
                                                    int* __restrict__ idx_out) {
    __shared__ float s_val[FPS_T / 32];   // 32 wave partials
    __shared__ int   s_idx[FPS_T / 32];
    __shared__ float s_last[3];

    const int b = blockIdx.x;
    const int t = threadIdx.x;
    const int lane = t & 31;
    const int w = t >> 5;
    const float* pb = p + (size_t)b * NPTS * 3;

    float dist[FPS_PPT];
#pragma unroll
    for (int j = 0; j < FPS_PPT; ++j) dist[j] = 1e10f;

    if (t == 0) {
        idx_out[b * MGRP] = 0;
        s_last[0] = pb[0]; s_last[1] = pb[1]; s_last[2] = pb[2];
    }
    __syncthreads();

    for (int it = 1; it < MGRP; ++it) {
        const float lx = s_last[0], ly = s_last[1], lz = s_last[2];
        float best = -1.0f;
        int bi = 0;
#pragma unroll
        for (int j = 0; j < FPS_PPT; ++j) {
            const int n = t + j * FPS_T;
            const float dx = pb[n * 3 + 0] - lx;
            const float dy = pb[n * 3 + 1] - ly;
            const float dz = pb[n * 3 + 2] - lz;
            const float d = dx * dx + dy * dy + dz * dz;
            const float dd = fminf(dist[j], d);
            dist[j] = dd;
            if (dd > best) { best = dd; bi = n; }   // strict > keeps lowest index
        }
        // wave32 argmax reduce (tie -> lowest index, matching jnp.argmax)
#pragma unroll
        for (int off = 16; off > 0; off >>= 1) {
            const float ov = __shfl_down(best, off, 32);
            const int   oi = __shfl_down(bi, off, 32);
            if (ov > best || (ov == best && oi < bi)) { best = ov; bi = oi; }
        }
        if (lane == 0) { s_val[w] = best; s_idx[w] = bi; }
        __syncthreads();
        if (w == 0) {
            best = s_val[lane];   // exactly 32 partials for a full wave32 reduce
            bi   = s_idx[lane];
#pragma unroll
            for (int off = 16; off > 0; off >>= 1) {
                const float ov = __shfl_down(best, off, 32);
                const int   oi = __shfl_down(bi, off, 32);
                if (ov > best || (ov == best && oi < bi)) { best = ov; bi = oi; }
            }
            if (lane == 0) {
                idx_out[b * MGRP + it] = bi;
                s_last[0] = pb[bi * 3 + 0];
                s_last[1] = pb[bi * 3 + 1];
                s_last[2] = pb[bi * 3 + 2];
            }
        }
        __syncthreads();
    }
}

// ---------------------------------------------------------------------------
// Kernel 2: ball query via V_WMMA_F32_16X16X4_F32.
// One wave handles 16 centers; each chunk of 16 points is one WMMA producing
// the 16x16 tile of (|c|^2 - 2 c.p); add |p|^2 per column, compare vs r^2.
// Software-pipelined point loads; per-center counts in registers.
// ---------------------------------------------------------------------------
__global__ void __launch_bounds__(256) ballq_kernel(const float* __restrict__ p,
                                                    const int* __restrict__ idx,
                                                    int* __restrict__ nidx) {
    __shared__ int s_list[8][16 * KGRP];   // per-wave lists: 16 centers x 32 slots

    const int wave = threadIdx.x >> 5;
    const int lane = threadIdx.x & 31;
    const int gw = blockIdx.x * 8 + wave;  // 0..511
    const int b = gw >> 6;                 // 64 waves per batch
    const int m0 = (gw & 63) * 16;         // first of 16 centers
    const float* pb = p + (size_t)b * NPTS * 3;

    int* list = s_list[wave];
    int cnt[16];
#pragma unroll
    for (int c = 0; c < 16; ++c) cnt[c] = 0;

    // Build A (16x4, f32): row m = [-2cx, -2cy, -2cz, |c|^2]
    // Layout: lanes 0-15 hold K=0 (VGPR0) / K=1 (VGPR1); lanes 16-31 hold K=2 / K=3.
    const int mloc = lane & 15;
    const int ci = idx[b * MGRP + m0 + mloc];
    const float cx = pb[ci * 3 + 0];
    const float cy = pb[ci * 3 + 1];
    const float cz = pb[ci * 3 + 2];
    v2f A;
    if (lane < 16) { A.x = -2.0f * cx; A.y = -2.0f * cy; }
    else           { A.x = -2.0f * cz; A.y = cx * cx + cy * cy + cz * cz; }

    // Software pipeline: preload chunk 0, load chunk j+1 while computing chunk j.
    float px = pb[mloc * 3 + 0];
    float py = pb[mloc * 3 + 1];
    float pz = pb[mloc * 3 + 2];

    int filled = 0;
    for (int j = 0; j < NCHUNK; ++j) {
        const float cpx = px, cpy = py, cpz = pz;
        // Branch-free wrapped prefetch index (last iteration re-loads chunk 0).
        const int nn = (((j + 1) & (NCHUNK - 1)) * 16 + mloc) * 3;
        __builtin_prefetch(&pb[(((j + 8) & (NCHUNK - 1)) * 16 + mloc) * 3], 0, 3);
        px = pb[nn + 0];
        py = pb[nn + 1];
        pz = pb[nn + 2];

        const int n0 = j * 16;
        // B (4x16): col n = [px, py, pz, 1]; lanes 0-15 -> K=0/1, lanes 16-31 -> K=2/3
        v2f Bv;
        if (lane < 16) { Bv.x = cpx; Bv.y = cpy; }
        else           { Bv.x = cpz; Bv.y = 1.0f; }
        const float pp = cpx * cpx + cpy * cpy + cpz * cpz;

        v8f acc = {};
        acc = __builtin_amdgcn_wmma_f32_16x16x4_f32(
            false, A, false, Bv, (short)0, acc, false, false);

        // D layout: lanes 0-15: N=lane, M=r; lanes 16-31: N=lane-16, M=r+8
        unsigned int mask[8];
        unsigned int any = 0u;
#pragma unroll
        for (int r = 0; r < 8; ++r) {
            mask[r] = __builtin_amdgcn_ballot_w32((acc[r] + pp) < RAD2);
            any |= mask[r];
        }
        if (any != 0u) {   // rare path (density ~0.4%)
#pragma unroll
            for (int r = 0; r < 8; ++r) {
                unsigned int mlo = mask[r] & 0xFFFFu;   // center r,   points n0..n0+15
                unsigned int mhi = mask[r] >> 16;       // center r+8, same points
                while (mlo) {
                    const int bit = __builtin_ctz(mlo);
                    mlo &= mlo - 1u;
                    if (cnt[r] < KGRP) {
                        list[r * KGRP + cnt[r]] = n0 + bit;
                        if (++cnt[r] == KGRP) ++filled;
                    }
                }
                while (mhi) {
                    const int bit = __builtin_ctz(mhi);
                    mhi &= mhi - 1u;
                    if (cnt[r + 8] < KGRP) {
                        list[(r + 8) * KGRP + cnt[r + 8]] = n0 + bit;
                        if (++cnt[r + 8] == KGRP) ++filled;
                    }
                }
            }
            if (filled >= 16) break;   // all 16 centers full: early exit
        }
    }

    // Fill unfilled slots with first-found index (0 if none), write nidx.
#pragma unroll
    for (int c = 0; c < 16; ++c) {
        const int cc = cnt[c];
        const int first = (cc > 0) ? list[c * KGRP] : 0;
        const int v = (lane < cc) ? list[c * KGRP + lane] : first;
        nidx[((size_t)(b * MGRP + m0 + c)) * KGRP + lane] = v;
    }
}

// ---------------------------------------------------------------------------
// Kernel 3: gathers + all four outputs. One wave per (b, m); lane = slot.
// ---------------------------------------------------------------------------
__global__ void __launch_bounds__(256) gather_kernel(const float* __restrict__ p,
                                                     const float* __restrict__ x,
                                                     const int* __restrict__ idx,
                                                     const int* __restrict__ nidx,
                                                     float* __restrict__ out_gp,
                                                     float* __restrict__ out_cp,
                                                     float* __restrict__ out_fj,
                                                     float* __restrict__ out_cx) {
    const int wave = threadIdx.x >> 5;
    const int lane = threadIdx.x & 31;
    const int gw = blockIdx.x * 8 + wave;   // 0..8191
    const int b = gw >> 10;
    const int m = gw & (MGRP - 1);

    const float* pb = p + (size_t)b * NPTS * 3;
    const int ci = idx[b * MGRP + m];
    const int ni = nidx[((size_t)(b * MGRP + m)) * KGRP + lane];

    const float cx = pb[ci * 3 + 0];
    const float cy = pb[ci * 3 + 1];
    const float cz = pb[ci * 3 + 2];

    const float gx = pb[ni * 3 + 0] - cx;
    const float gy = pb[ni * 3 + 1] - cy;
    const float gz = pb[ni * 3 + 2] - cz;

    // grouped_p: [B, 3, M, K]
    out_gp[(((size_t)b * 3 + 0) * MGRP + m) * KGRP + lane] = gx;
    out_gp[(((size_t)b * 3 + 1) * MGRP + m) * KGRP + lane] = gy;
    out_gp[(((size_t)b * 3 + 2) * MGRP + m) * KGRP + lane] = gz;

    // center_p: [B, M, 3]
    if (lane < 3) out_cp[((size_t)b * MGRP + m) * 3 + lane] = pb[ci * 3 + lane];

    // fj: [B, C, M, K]; center_x: [B, C, M, 1]
    const float* xb = x + (size_t)b * NCH * NPTS;
#pragma unroll 8
    for (int c = 0; c < NCH; ++c) {
        const float v = xb[(size_t)c * NPTS + ni];
        out_fj[(((size_t)b * NCH + c) * MGRP + m) * KGRP + lane] = v;
        if (lane == 0)
            out_cx[((size_t)b * NCH + c) * MGRP + m] = xb[(size_t)c * NPTS + ci];
    }
}

// ---------------------------------------------------------------------------
extern "C" void kernel_launch(void* const* d_in, const int* in_sizes, int n_in,
                              void* d_out, int out_size, void* d_ws, size_t ws_size,
                              hipStream_t stream) {
    (void)in_sizes; (void)n_in; (void)out_size; (void)ws_size;
    const float* p = (const float*)d_in[0];   // [B, N, 3]
    const float* x = (const float*)d_in[1];   // [B, C, N]

    float* out = (float*)d_out;
    float* out_gp = out;                                          // B*3*M*K = 786432
    float* out_cp = out_gp + (size_t)BATCH * 3 * MGRP * KGRP;     // B*M*3   = 24576
    float* out_fj = out_cp + (size_t)BATCH * MGRP * 3;            // B*C*M*K = 16777216
    float* out_cx = out_fj + (size_t)BATCH * NCH * MGRP * KGRP;   // B*C*M   = 524288

    int* idx  = (int*)d_ws;                   // B*M ints       (32 KB)
    int* nidx = idx + (size_t)BATCH * MGRP;   // B*M*K ints     (1 MB)

    fps_kernel<<<BATCH, FPS_T, 0, stream>>>(p, idx);
    ballq_kernel<<<(BATCH * MGRP / 16) / 8, 256, 0, stream>>>(p, idx, nidx);
    gather_kernel<<<(BATCH * MGRP) / 8, 256, 0, stream>>>(p, x, idx, nidx,
                                                          out_gp, out_cp, out_fj, out_cx);
}